// LongformerPersonalizedClsHead_50646254354614
// MI455X (gfx1250) — compile-verified
//
#include <hip/hip_runtime.h>
#include <math.h>

// Problem constants
#define B_  256
#define S_  512
#define H_  768
#define U_  100
#define L_  2
#define K_  (U_ * H_)        // 76800

// GEMM1 tiling
#define KS        8          // split-K factor
#define MT        32         // rows per workgroup tile
#define NT        128        // cols per workgroup tile
#define KT        32         // K per LDS stage
#define NTHREADS  256        // 8 waves (wave32)
#define KCHUNK    (K_ / KS)  // 9600
#define NITER     (KCHUNK / KT) // 300
#define A_STRIDE  36         // padded LDS row stride for A tile (bank-conflict-free)

typedef __attribute__((ext_vector_type(2))) float        v2f;
typedef __attribute__((ext_vector_type(4))) float        v4f;
typedef __attribute__((ext_vector_type(8))) float        v8f;
typedef __attribute__((ext_vector_type(4))) unsigned int u32x4;
typedef __attribute__((ext_vector_type(4))) int          i32x4;
typedef __attribute__((ext_vector_type(8))) int          i32x8;

// -----------------------------------------------------------------------------
// Kernel 1: split-K WMMA GEMM.  P[kz][b][n] = sum_{k in chunk kz} XM[b,k] * W[k,n]
// where XM[b, u*H+h] = mask[b,u] * x[b,h] is generated on the fly.
// B tiles are staged into LDS by the Tensor Data Mover (double-buffered).
// -----------------------------------------------------------------------------
__global__ __launch_bounds__(NTHREADS)
void gemm1_kernel(const float* __restrict__ HS,    // (B, S, H); we read [:,0,:]
                  const float* __restrict__ mask,  // (B, U)
                  const float* __restrict__ W,     // (U*H, H) row-major view of dense_W
                  float* __restrict__ P)           // (KS, B, H) partials
{
    __shared__ __align__(16) float maskS[MT * U_];          // 32x100
    __shared__ __align__(16) float As[2][MT * A_STRIDE];    // double-buffered A tile
    __shared__ __align__(16) float Bs[2][KT * NT];          // double-buffered B tile

    const int tid = threadIdx.x;
    int blk = blockIdx.x;                 // (mt * 6 + nt) * KS + kz
    const int kz = blk % KS;  blk /= KS;
    const int nt = blk % (H_ / NT); blk /= (H_ / NT);
    const int mt = blk;                   // 0..7
    const int m0 = mt * MT;
    const int n0 = nt * NT;
    const int kbase0 = kz * KCHUNK;

    const int wave = tid >> 5;
    const int lane = tid & 31;
    const int lo   = lane & 15;
    const int hi   = lane >> 4;
    const int wm   = (wave >> 2) << 4;  // 0 or 16
    const int wn   = (wave & 3) << 5;   // 0,32,64,96

    // --- TDM descriptor issue: 32x128 fp32 tile of W starting at row kb -----
    const unsigned ldsB[2] = { (unsigned)(unsigned long long)(void*)&Bs[0][0],
                               (unsigned)(unsigned long long)(void*)&Bs[1][0] };
    auto tdm_issue = [&](int buf, int kb) {
        const unsigned long long ga =
            (unsigned long long)(const void*)(W + (size_t)kb * H_ + n0);
        u32x4 g0;
        g0.x = 1u;                                    // count=1, user descriptor
        g0.y = ldsB[buf];                             // LDS byte address
        g0.z = (unsigned)ga;                          // global_addr[31:0]
        g0.w = (unsigned)((ga >> 32) & 0x1FFFFFFull)  // global_addr[56:32]
             | (2u << 30);                            // type = 2 ("image")
        i32x8 g1;
        g1[0] = (int)(2u << 16);                      // data_size = 4B
        g1[1] = (int)((unsigned)H_ << 16);            // tensor_dim0 = 768
        g1[2] = (int)((unsigned)KT << 16);            // tensor_dim1 = 32 (tile rows)
        g1[3] = (int)((unsigned)NT << 16);            // tile_dim0 = 128
        g1[4] = (int)KT;                              // tile_dim1 = 32
        g1[5] = (int)H_;                              // tensor_dim0_stride = 768
        g1[6] = 0;
        g1[7] = 0;
        const i32x4 z4 = {0, 0, 0, 0};
        const i32x8 z8 = {0, 0, 0, 0, 0, 0, 0, 0};
        __builtin_amdgcn_tensor_load_to_lds(g0, g1, z4, z4, z8, 0);
    };

    // --- mask tile -> LDS ----------------------------------------------------
    for (int i = tid; i < MT * U_; i += NTHREADS) {
        const int m = i / U_;
        const int u = i - m * U_;
        maskS[i] = mask[(m0 + m) * U_ + u];
    }
    __syncthreads();          // maskS ready (A staging reads it)

    // --- incremental A staging state ----------------------------------------
    const int amr  = tid >> 3;       // A row 0..31
    const int akk4 = (tid & 7) << 2; // A k-offset 0,4,...,28 (multiple of 4)
    unsigned ak = (unsigned)(kbase0 + akk4);
    unsigned au = ak / (unsigned)H_;
    unsigned ah = ak - au * (unsigned)H_;   // multiple of 4 -> the 4-elem group
                                            // never straddles a u boundary
    float mval = maskS[amr * U_ + au];
    const float* xrow = HS + (size_t)(m0 + amr) * (S_ * H_);   // token 0 row

    auto stageA = [&](float* dst) {
        const v4f xv = *(const v4f*)(xrow + ah);
        *(v4f*)(dst + amr * A_STRIDE + akk4) = xv * mval;
        ah += KT;
        if (ah >= (unsigned)H_) {
            ah -= (unsigned)H_;
            ++au;
            if (au < (unsigned)U_) mval = maskS[amr * U_ + au];
        }
    };

    // --- prologue: A(0) staged; B(0), B(1) in flight via TDM -----------------
    stageA(&As[0][0]);
    if (wave == 0) {
        tdm_issue(0, kbase0);
        tdm_issue(1, kbase0 + KT);
        __builtin_amdgcn_s_wait_tensorcnt(1);   // B(0) resident (in-order)
    }
    __syncthreads();

    v8f c0 = {};
    v8f c1 = {};

    for (int it = 0; it < NITER; ++it) {
        const int cur = it & 1;
        const int nxt = cur ^ 1;
        const bool more1 = (it + 1) < NITER;
        const bool more2 = (it + 2) < NITER;

        if (more1) stageA(&As[nxt][0]);   // writes only the nxt buffer

        // 8 WMMA K-steps over the staged 32-deep tile.
        const float* arow = &As[cur][0] + (wm + lo) * A_STRIDE;
        const float* bt   = &Bs[cur][0];
        #pragma unroll
        for (int kk = 0; kk < KT; kk += 4) {
            const int krow = kk + (hi << 1);
            // A fragment: VGPR0 = A[m][krow], VGPR1 = A[m][krow+1]
            v2f a = *(const v2f*)(arow + krow);
            // B fragments: VGPR0 = B[krow][n], VGPR1 = B[krow+1][n]
            v2f b0; b0.x = bt[krow * NT + wn + lo];
                    b0.y = bt[(krow + 1) * NT + wn + lo];
            v2f b1; b1.x = bt[krow * NT + wn + 16 + lo];
                    b1.y = bt[(krow + 1) * NT + wn + 16 + lo];
            c0 = __builtin_amdgcn_wmma_f32_16x16x4_f32(false, a, false, b0,
                                                       (short)0, c0, false, false);
            c1 = __builtin_amdgcn_wmma_f32_16x16x4_f32(false, a, false, b1,
                                                       (short)0, c1, false, false);
        }

        __syncthreads();      // everyone done reading Bs[cur] / As[cur]
        if (wave == 0) {
            if (more2) {
                tdm_issue(cur, kbase0 + (it + 2) * KT);  // overwrite cur
                __builtin_amdgcn_s_wait_tensorcnt(1);    // B(it+1) resident
            } else if (more1) {
                __builtin_amdgcn_s_wait_tensorcnt(0);    // last pending tile
            }
        }
        __syncthreads();      // B(it+1) published to all waves
    }

    // Write partial tile. C/D layout: vgpr g holds M=g (lanes 0-15) / M=g+8 (16-31),
    // lane&15 selects N.
    float* Pp = P + ((size_t)kz * B_ + m0) * H_ + n0;
    #pragma unroll
    for (int g = 0; g < 8; ++g) {
        const int row = wm + g + (hi << 3);
        Pp[(size_t)row * H_ + wn + lo]      = c0[g];
        Pp[(size_t)row * H_ + wn + 16 + lo] = c1[g];
    }
}

// -----------------------------------------------------------------------------
// Kernel 2: reduce split-K partials, add mask-mixed dense bias, tanh.
// -----------------------------------------------------------------------------
__global__ __launch_bounds__(256)
void bias_tanh_kernel(const float* __restrict__ mask,
                      const float* __restrict__ db,   // (U, H)
                      const float* __restrict__ P,    // (KS, B, H)
                      float* __restrict__ Hout)       // (B, H)
{
    const int idx = blockIdx.x * 256 + threadIdx.x;   // < B*H
    const int b = idx / H_;
    const int n = idx - b * H_;
    float acc = 0.0f;
    const float* mrow = mask + b * U_;
    #pragma unroll 4
    for (int u = 0; u < U_; ++u)
        acc += mrow[u] * db[u * H_ + n];
    #pragma unroll
    for (int kz = 0; kz < KS; ++kz)
        acc += P[((size_t)kz * B_ + b) * H_ + n];
    Hout[idx] = tanhf(acc);
}

// -----------------------------------------------------------------------------
// Kernel 3: out[b,l] = sum_k (mask[b,k/H]*h[b,k%H]) * Wo_flat[k,l] + bo_mix[b,l]
// (L=2, K=76800) — one block per batch row, tree reduction.
// -----------------------------------------------------------------------------
__global__ __launch_bounds__(256)
void out_kernel(const float* __restrict__ mask,
                const float* __restrict__ Hin,   // (B, H)
                const float* __restrict__ Wo,    // (U*H, 2)
                const float* __restrict__ bo,    // (U, 2)
                float* __restrict__ out)         // (B, 2)
{
    __shared__ float maskS[U_];
    __shared__ float red0[256];
    __shared__ float red1[256];
    const int b = blockIdx.x;
    const int t = threadIdx.x;
    if (t < U_) maskS[t] = mask[b * U_ + t];
    __syncthreads();

    const float* hrow = Hin + b * H_;
    float a0 = 0.0f, a1 = 0.0f;
    for (int k = t; k < K_; k += 256) {
        const unsigned u = (unsigned)k / (unsigned)H_;
        const unsigned h = (unsigned)k - u * (unsigned)H_;
        const float hv = maskS[u] * hrow[h];
        const v2f w = *(const v2f*)(Wo + (size_t)k * 2);
        a0 += hv * w.x;
        a1 += hv * w.y;
    }
    red0[t] = a0;
    red1[t] = a1;
    __syncthreads();
    for (int s = 128; s > 0; s >>= 1) {
        if (t < s) { red0[t] += red0[t + s]; red1[t] += red1[t + s]; }
        __syncthreads();
    }
    if (t == 0) {
        float bb0 = 0.0f, bb1 = 0.0f;
        for (int u = 0; u < U_; ++u) {
            bb0 += maskS[u] * bo[u * 2 + 0];
            bb1 += maskS[u] * bo[u * 2 + 1];
        }
        out[b * 2 + 0] = red0[0] + bb0;
        out[b * 2 + 1] = red1[0] + bb1;
    }
}

// -----------------------------------------------------------------------------
extern "C" void kernel_launch(void* const* d_in, const int* in_sizes, int n_in,
                              void* d_out, int out_size, void* d_ws, size_t ws_size,
                              hipStream_t stream) {
    const float* HS   = (const float*)d_in[0];  // hidden_states (B,S,H)
    const float* mask = (const float*)d_in[1];  // user_mask     (B,U)
    const float* dW   = (const float*)d_in[2];  // dense_W       (U,H,H)
    const float* db   = (const float*)d_in[3];  // dense_b       (U,H)
    const float* Wo   = (const float*)d_in[4];  // out_proj_W    (U,H,L)
    const float* bo   = (const float*)d_in[5];  // out_proj_b    (U,L)
    float* out = (float*)d_out;

    float* P    = (float*)d_ws;                          // KS*B*H floats (6 MB)
    float* Hbuf = P + (size_t)KS * B_ * H_;              // B*H floats (0.75 MB)

    const int grid1 = (B_ / MT) * (H_ / NT) * KS;        // 8*6*8 = 384 blocks
    gemm1_kernel<<<grid1, NTHREADS, 0, stream>>>(HS, mask, dW, P);
    bias_tanh_kernel<<<(B_ * H_) / 256, 256, 0, stream>>>(mask, db, P, Hbuf);
    out_kernel<<<B_, 256, 0, stream>>>(mask, Hbuf, Wo, bo, out);
}